// SparseGATLayer_26414048870697
// MI455X (gfx1250) — compile-verified
//
#include <hip/hip_runtime.h>

typedef __attribute__((ext_vector_type(2))) float v2f;
typedef __attribute__((ext_vector_type(8))) float v8f;

#define N_NODES 100000
#define N_EDGES 1000000
#define D 64
#define LEAKY 0.2f
#define EPSV 1e-10f
#define NEG_INF (-3.402823466e38f)

// ---------------------------------------------------------------------------
// K0: u_src[k] = sum_o W[o][k] * a[o];  u_dst[k] = sum_o W[o][k] * a[64+o]
// one block of 64 threads
// ---------------------------------------------------------------------------
__global__ __launch_bounds__(64) void compute_u(const float* __restrict__ W,
                                                const float* __restrict__ a,
                                                float* __restrict__ u) {
    int k = threadIdx.x;
    float us = 0.f, ud = 0.f;
    #pragma unroll 8
    for (int o = 0; o < D; ++o) {
        float w = W[o * D + k];
        us += w * a[o];
        ud += w * a[D + o];
    }
    u[k]     = us;
    u[D + k] = ud;
}

// ---------------------------------------------------------------------------
// K1: Wh = h @ W^T via V_WMMA_F32_16X16X4_F32.
// Block = 128 threads = 4 waves; wave w computes 16x16 tile (rows 16*bid, cols 16*w).
// K=64 -> 16 wmma steps of K=4, fully unrolled.
// A (16x4): lane m=L%16, half=L/16; a.{x,y} = h[m][kk+2*half +{0,1}]
// B (4x16): B[k][n] = W[n][k]; b.{x,y} = W[n][kk+2*half +{0,1}], n=L%16
// D: acc[r] = Wh[r + 8*half][n]
// ---------------------------------------------------------------------------
__global__ __launch_bounds__(128) void wmma_gemm(const float* __restrict__ h,
                                                 const float* __restrict__ W,
                                                 float* __restrict__ Wh) {
    const int wave = threadIdx.x >> 5;      // 0..3 -> N tile
    const int lane = threadIdx.x & 31;
    const int m0   = blockIdx.x * 16;
    const int n0   = wave * 16;
    const int m    = lane & 15;
    const int half = lane >> 4;

    const float* __restrict__ hrow = h + (size_t)(m0 + m) * D;
    const float* __restrict__ wrow = W + (size_t)(n0 + m) * D;   // n == m index pattern

    v8f acc = {};
    #pragma unroll
    for (int kk = 0; kk < D; kk += 4) {
        v2f av, bv;
        av.x = hrow[kk + 2 * half];
        av.y = hrow[kk + 2 * half + 1];
        bv.x = wrow[kk + 2 * half];
        bv.y = wrow[kk + 2 * half + 1];
        acc = __builtin_amdgcn_wmma_f32_16x16x4_f32(
            /*neg_a=*/false, av, /*neg_b=*/false, bv,
            /*c_mod=*/(short)0, acc, /*reuse_a=*/false, /*reuse_b=*/false);
    }

    #pragma unroll
    for (int r = 0; r < 8; ++r) {
        int mr = m0 + r + 8 * half;
        int nc = n0 + m;
        Wh[(size_t)mr * D + nc] = acc[r];
    }
}

// ---------------------------------------------------------------------------
// K2: per-node scores, one wave per node: s_src[i] = h[i].u_src, s_dst[i]=h[i].u_dst
// ---------------------------------------------------------------------------
__global__ __launch_bounds__(256) void node_scores(const float* __restrict__ h,
                                                   const float* __restrict__ u,
                                                   float* __restrict__ s_src,
                                                   float* __restrict__ s_dst) {
    int node = blockIdx.x * 8 + (threadIdx.x >> 5);
    int lane = threadIdx.x & 31;
    float2 hv = ((const float2*)(h + (size_t)node * D))[lane];
    float2 us = ((const float2*)u)[lane];
    float2 ud = ((const float2*)(u + D))[lane];
    float ss = hv.x * us.x + hv.y * us.y;
    float sd = hv.x * ud.x + hv.y * ud.y;
    #pragma unroll
    for (int off = 16; off > 0; off >>= 1) {
        ss += __shfl_down(ss, off, 32);
        sd += __shfl_down(sd, off, 32);
    }
    if (lane == 0) { s_src[node] = ss; s_dst[node] = sd; }
}

// ---------------------------------------------------------------------------
// K3: zero fill
// ---------------------------------------------------------------------------
__global__ void zerof(float* __restrict__ p, int n) {
    int i = blockIdx.x * blockDim.x + threadIdx.x;
    if (i < n) p[i] = 0.0f;
}

// ---------------------------------------------------------------------------
// K4: per-edge raw score + leaky relu; per-block max
// ---------------------------------------------------------------------------
__global__ __launch_bounds__(256) void edge_scores(const int* __restrict__ rows,
                                                   const int* __restrict__ cols,
                                                   const float* __restrict__ s_src,
                                                   const float* __restrict__ s_dst,
                                                   float* __restrict__ e,
                                                   float* __restrict__ blockmax) {
    __shared__ float smax[256];
    int k = blockIdx.x * 256 + threadIdx.x;
    float v = NEG_INF;
    if (k < N_EDGES) {
        float x = s_src[rows[k]] + s_dst[cols[k]];
        x = (x > 0.0f) ? x : LEAKY * x;
        e[k] = x;
        v = x;
    }
    smax[threadIdx.x] = v;
    __syncthreads();
    #pragma unroll
    for (int s = 128; s > 0; s >>= 1) {
        if (threadIdx.x < s)
            smax[threadIdx.x] = fmaxf(smax[threadIdx.x], smax[threadIdx.x + s]);
        __syncthreads();
    }
    if (threadIdx.x == 0) blockmax[blockIdx.x] = smax[0];
}

// ---------------------------------------------------------------------------
// K5: reduce block maxima to one value (single block)
// ---------------------------------------------------------------------------
__global__ __launch_bounds__(1024) void reduce_max(const float* __restrict__ blockmax,
                                                   int n, float* __restrict__ gmax) {
    __shared__ float sm[1024];
    float v = NEG_INF;
    for (int i = threadIdx.x; i < n; i += 1024) v = fmaxf(v, blockmax[i]);
    sm[threadIdx.x] = v;
    __syncthreads();
    #pragma unroll
    for (int s = 512; s > 0; s >>= 1) {
        if (threadIdx.x < s) sm[threadIdx.x] = fmaxf(sm[threadIdx.x], sm[threadIdx.x + s]);
        __syncthreads();
    }
    if (threadIdx.x == 0) *gmax = sm[0];
}

// ---------------------------------------------------------------------------
// K6: exp(e - max), segment-sum into rows via f32 atomics
// ---------------------------------------------------------------------------
__global__ __launch_bounds__(256) void edge_exp(const int* __restrict__ rows,
                                                const float* __restrict__ e,
                                                const float* __restrict__ gmax,
                                                float* __restrict__ exp_e,
                                                float* __restrict__ sum_exp) {
    int k = blockIdx.x * 256 + threadIdx.x;
    if (k < N_EDGES) {
        float x = __expf(e[k] - *gmax);
        exp_e[k] = x;
        atomicAdd(&sum_exp[rows[k]], x);
    }
}

// ---------------------------------------------------------------------------
// K7: aggregation, one wave per edge; lane handles 2 dims (float2 gather),
// scatter-add with global f32 atomics (L2-resident).
// ---------------------------------------------------------------------------
__global__ __launch_bounds__(256) void aggregate(const int* __restrict__ rows,
                                                 const int* __restrict__ cols,
                                                 const float* __restrict__ exp_e,
                                                 const float* __restrict__ sum_exp,
                                                 const float* __restrict__ Wh,
                                                 float* __restrict__ out) {
    int k    = blockIdx.x * 8 + (threadIdx.x >> 5);
    int lane = threadIdx.x & 31;
    int r = rows[k];
    int c = cols[k];
    float alpha = exp_e[k] / (sum_exp[r] + EPSV);
    float2 wv = ((const float2*)(Wh + (size_t)c * D))[lane];
    float* dst = out + (size_t)r * D + 2 * lane;
    atomicAdd(dst,     alpha * wv.x);
    atomicAdd(dst + 1, alpha * wv.y);
}

// ---------------------------------------------------------------------------
// K8: ELU in place
// ---------------------------------------------------------------------------
__global__ void elu_k(float* __restrict__ out, int n) {
    int i = blockIdx.x * blockDim.x + threadIdx.x;
    if (i < n) {
        float x = out[i];
        out[i] = (x > 0.0f) ? x : (__expf(x) - 1.0f);
    }
}

// ---------------------------------------------------------------------------
extern "C" void kernel_launch(void* const* d_in, const int* in_sizes, int n_in,
                              void* d_out, int out_size, void* d_ws, size_t ws_size,
                              hipStream_t stream) {
    const float* h    = (const float*)d_in[0];
    const float* W    = (const float*)d_in[1];
    const float* a    = (const float*)d_in[2];
    const int*   rows = (const int*)d_in[3];
    const int*   cols = (const int*)d_in[4];
    float* out = (float*)d_out;

    // workspace layout (floats)
    float* ws       = (float*)d_ws;
    float* u        = ws;                       // 128   (+pad to 256)
    float* Wh       = ws + 256;                 // 6,400,000
    float* s_src    = Wh + (size_t)N_NODES * D; // 100,000
    float* s_dst    = s_src + N_NODES;          // 100,000
    float* e        = s_dst + N_NODES;          // 1,000,000
    float* exp_e    = e + N_EDGES;              // 1,000,000
    float* sum_exp  = exp_e + N_EDGES;          // 100,000
    float* blockmax = sum_exp + N_NODES;        // 3907 (+pad 4096)
    float* gmax     = blockmax + 4096;          // 1

    const int EB = (N_EDGES + 255) / 256;       // 3907 edge blocks

    // K0: u vectors
    compute_u<<<1, 64, 0, stream>>>(W, a, u);

    // K1: Wh = h @ W^T  (WMMA f32 16x16x4, 16 steps)
    wmma_gemm<<<N_NODES / 16, 128, 0, stream>>>(h, W, Wh);

    // K2: per-node scores
    node_scores<<<N_NODES / 8, 256, 0, stream>>>(h, u, s_src, s_dst);

    // K3: zero accumulators (sum_exp and output)
    zerof<<<(N_NODES + 255) / 256, 256, 0, stream>>>(sum_exp, N_NODES);
    zerof<<<(N_NODES * D + 255) / 256, 256, 0, stream>>>(out, N_NODES * D);

    // K4: edge scores + block max
    edge_scores<<<EB, 256, 0, stream>>>(rows, cols, s_src, s_dst, e, blockmax);

    // K5: global max
    reduce_max<<<1, 1024, 0, stream>>>(blockmax, EB, gmax);

    // K6: exp + segment sum
    edge_exp<<<EB, 256, 0, stream>>>(rows, e, gmax, exp_e, sum_exp);

    // K7: aggregation (wave per edge)
    aggregate<<<N_EDGES / 8, 256, 0, stream>>>(rows, cols, exp_e, sum_exp, Wh, out);

    // K8: ELU
    elu_k<<<(N_NODES * D + 255) / 256, 256, 0, stream>>>(out, N_NODES * D);
}